// VectorsFromBbox_83760452206672
// MI455X (gfx1250) — compile-verified
//
#include <hip/hip_runtime.h>
#include <math.h>

// encoded [B,C,H,W] f32, bboxes [B,N,4] int64 (x,y,w,h), out [B,C,N,1] f32.
#define BB 8
#define CC 128
#define HH 96
#define WW 96
#define NN 16
#define PLANE (HH * WW)          // 9216 floats = 36 KB per (b,c) plane

typedef int v4i __attribute__((vector_size(16)));        // 16-byte payload type
typedef __attribute__((address_space(1))) v4i GV4;       // global v4i
typedef __attribute__((address_space(3))) v4i LV4;       // LDS v4i
typedef __attribute__((address_space(3))) void LVoid;    // LDS void

// Async DMA of 16 bytes global -> LDS (gfx1250 GLOBAL_LOAD_ASYNC_TO_LDS_B128,
// tracked by ASYNCcnt). Builtin confirmed present; param element type is v4i.
// We pass AS1/AS3-qualified pointers: exact match if params are AS-qualified,
// implicit AS->generic conversion if they are generic.
__device__ __forceinline__ void async_copy_b128(const float* gptr, float* lptr) {
#if __has_builtin(__builtin_amdgcn_global_load_async_to_lds_b128)
  GV4* g = (GV4*)(__attribute__((address_space(1))) void*)(void*)gptr;
  LV4* l = (LV4*)(LVoid*)lptr;
  __builtin_amdgcn_global_load_async_to_lds_b128(g, l, /*imm_offset=*/0, /*cpol=*/0);
#else
  unsigned lds_addr = (unsigned)(__SIZE_TYPE__)(LVoid*)lptr;  // LDS byte addr
  unsigned long long ga = (unsigned long long)(__SIZE_TYPE__)gptr;
  asm volatile("global_load_async_to_lds_b128 %0, %1, off"
               :: "v"(lds_addr), "v"(ga) : "memory");
#endif
}

__device__ __forceinline__ void wait_asynccnt0() {
#if __has_builtin(__builtin_amdgcn_s_wait_asynccnt)
  __builtin_amdgcn_s_wait_asynccnt(0);
#else
  asm volatile("s_wait_asynccnt 0x0" ::: "memory");
#endif
}

// One workgroup (256 threads = 8 wave32) per (b,c) plane:
//   1) async-DMA the whole 96x96 plane into LDS (each input byte read once),
//   2) s_wait_asynccnt 0 + barrier,
//   3) each wave reduces 2 boxes from LDS with a shfl_xor max tree.
__global__ __launch_bounds__(256) void bbox_maxpool_lds_kernel(
    const float* __restrict__ enc,
    const long long* __restrict__ bboxes,
    float* __restrict__ out) {
  __shared__ float tile[PLANE];

  const int bc  = blockIdx.x;         // 0 .. B*C-1
  const int b   = bc >> 7;            // log2(CC) = 7
  const int tid = threadIdx.x;

  const float* plane = enc + (__SIZE_TYPE__)bc * PLANE;

  // Stage: 2304 x b128 chunks, 9 per thread. Issue all, then wait once.
  #pragma unroll
  for (int i = tid; i < PLANE / 4; i += 256) {
    async_copy_b128(plane + i * 4, &tile[i * 4]);
  }
  wait_asynccnt0();
  __syncthreads();

  const int wv   = tid >> 5;          // wave id 0..7
  const int lane = tid & 31;

  for (int n = wv; n < NN; n += 8) {
    // Box coords are wave-uniform.
    const long long* bb = bboxes + ((__SIZE_TYPE__)(b * NN + n) * 4);
    const int x = (int)bb[0];
    const int y = (int)bb[1];
    const int w = (int)bb[2];
    const int h = (int)bb[3];

    const float* rbase = &tile[y * WW + x];
    const int area = w * h;
    float m = -INFINITY;
    for (int i = lane; i < area; i += 32) {
      const int r  = i / w;           // w is wave-uniform
      const int cl = i - r * w;
      m = fmaxf(m, rbase[r * WW + cl]);   // consecutive lanes -> consecutive banks
    }

    // wave32 max-reduction (CDNA5 is wave32-only).
    #pragma unroll
    for (int off = 16; off > 0; off >>= 1) {
      m = fmaxf(m, __shfl_xor(m, off, 32));
    }

    if (lane == 0) {
      // out layout [B, C, N, 1]; bc*NN + n == ((b*CC + c)*NN + n)
      out[(__SIZE_TYPE__)bc * NN + n] = m;
    }
  }
}

extern "C" void kernel_launch(void* const* d_in, const int* in_sizes, int n_in,
                              void* d_out, int out_size, void* d_ws, size_t ws_size,
                              hipStream_t stream) {
  (void)in_sizes; (void)n_in; (void)d_ws; (void)ws_size; (void)out_size;
  const float*     enc    = (const float*)d_in[0];
  const long long* bboxes = (const long long*)d_in[1];
  float*           out    = (float*)d_out;

  const int grid  = BB * CC;   // 1024 workgroups, one per (b,c) plane
  const int block = 256;       // 8 wave32
  bbox_maxpool_lds_kernel<<<grid, block, 0, stream>>>(enc, bboxes, out);
}